// SpectralConv2d_56891136803251
// MI455X (gfx1250) — compile-verified
//
#include <hip/hip_runtime.h>

typedef float v2f __attribute__((ext_vector_type(2)));
typedef float v8f __attribute__((ext_vector_type(8)));

#define NX 256
#define NT 256
#define NC 32
#define NB 16
#define NIMG (NB * NC)   // 512 images per pass (batch x channel)
#define NSLOT 32         // 4 bands * 8 kx rows
#define NKT 16
#define TWO_PI 6.28318530717958647692f

// kx row index for each of the 32 "slots" (band-major, matching w[0..3])
__device__ __forceinline__ int kx_of_slot(int s) {
    const int band = s >> 3;
    const int kk = s & 7;
    if (band == 0) return kk;          // pos-low  rows [0,8)
    if (band == 1) return 50 + kk;     // pos-mid  rows [50,58)
    if (band == 2) return 248 + kk;    // neg-low  rows [248,256)
    return 198 + kk;                   // neg-mid  rows [198,206)
}

// ---------------------------------------------------------------------------
// Init: build the two constant DFT matrices in workspace.
//   Afwd[64][256]: rows 0..31 = cos(2*pi*kx*x/256), rows 32..63 = -sin(...)
//   Binv[32][256]: irfft-over-t basis with 1/N and Hermitian doubling folded in
// ---------------------------------------------------------------------------
__global__ __launch_bounds__(256) void init_mats(float* __restrict__ Afwd,
                                                 float* __restrict__ Binv) {
    const int idx = blockIdx.x * blockDim.x + threadIdx.x;
    if (idx < 64 * 256) {
        const int r = idx >> 8;
        const int x = idx & 255;
        const int kx = kx_of_slot(r & 31);
        float s, c;
        sincosf(TWO_PI * (float)((kx * x) & 255) * (1.0f / 256.0f), &s, &c);
        Afwd[idx] = (r < 32) ? c : -s;
    }
    const int j = idx - 64 * 256;
    if (j >= 0 && j < 32 * 256) {
        const int k = j >> 8;
        const int t = j & 255;
        float v;
        if (k == 0) {
            v = 1.0f / 256.0f;                       // DC (real part only)
        } else if (k < 16) {
            float s, c;
            sincosf(TWO_PI * (float)((k * t) & 255) * (1.0f / 256.0f), &s, &c);
            v = (2.0f / 256.0f) * c;                 // Re coefficient
        } else if (k == 16) {
            v = 0.0f;                                // Im of DC ignored by irfft
        } else {
            float s, c;
            sincosf(TWO_PI * (float)(((k - 16) * t) & 255) * (1.0f / 256.0f), &s, &c);
            v = -(2.0f / 256.0f) * s;                // Im coefficient
        }
        Binv[j] = v;
    }
}

// ---------------------------------------------------------------------------
// F1: per image, Y[64][256] = Afwd[64][256] x X[256][256]  (fp32 WMMA GEMM)
// One wave per (image, n-tile): computes all 4 m-tiles (4 accumulators) so
// each B (X) fragment is loaded once and reused by 4 WMMAs -> X read exactly
// once from HBM.  Afwd is 64 KB and L2-resident.
// ---------------------------------------------------------------------------
__global__ __launch_bounds__(256) void f1_gemm(const float* __restrict__ x,
                                               const float* __restrict__ Afwd,
                                               float* __restrict__ Y) {
    const int gid  = blockIdx.x * blockDim.x + threadIdx.x;
    const int wave = gid >> 5;               // 512 images * 16 n-tiles = 8192 waves
    const int lane = threadIdx.x & 31;
    const int img  = wave >> 4;
    const int n0   = (wave & 15) << 4;

    const float* __restrict__ X = x + (size_t)img * (NX * NT);
    const int mrow  = lane & 15;             // M (for A) / N (for B) within tile
    const int khalf = (lane >> 4) << 1;      // lanes 16..31 hold K+2,K+3

    v8f acc[4] = {};
#pragma unroll 4
    for (int k0 = 0; k0 < NX; k0 += 4) {
        if ((k0 & 15) == 0) {
            // prefetch the k-panel 16 rows ahead (speculative; OOB is dropped)
            __builtin_prefetch(&X[(size_t)(k0 + 16 + khalf) * NT + n0 + mrow], 0, 1);
        }
        v2f b;
        b[0] = X[(size_t)(k0 + khalf)     * NT + n0 + mrow];
        b[1] = X[(size_t)(k0 + khalf + 1) * NT + n0 + mrow];
#pragma unroll
        for (int m = 0; m < 4; ++m) {
            const float* __restrict__ arow = Afwd + (m * 16 + mrow) * NX;
            v2f a;
            a[0] = arow[k0 + khalf];
            a[1] = arow[k0 + khalf + 1];
            acc[m] = __builtin_amdgcn_wmma_f32_16x16x4_f32(
                false, a, false, b, (short)0, acc[m], false, false);
        }
    }

    float* __restrict__ Yt = Y + (size_t)img * (64 * NT);
    const int half = lane >> 4;
#pragma unroll
    for (int m = 0; m < 4; ++m)
#pragma unroll
        for (int r = 0; r < 8; ++r)
            Yt[(size_t)(m * 16 + r + 8 * half) * NT + n0 + (lane & 15)] = acc[m][r];
}

// ---------------------------------------------------------------------------
// F2: t-DFT to kt = 0..15.  Z[img][slot][kt] = sum_t Y[slot][t] * e^{-i 2pi kt t/256}
// One thread per (img, slot, kt) -> 262144 threads.
// ---------------------------------------------------------------------------
__global__ __launch_bounds__(256) void f2_dft(const float* __restrict__ Y,
                                              float* __restrict__ Zr,
                                              float* __restrict__ Zi) {
    const int idx  = blockIdx.x * blockDim.x + threadIdx.x;
    const int kt   = idx & 15;
    const int slot = (idx >> 4) & 31;
    const int img  = idx >> 9;
    const float* __restrict__ yr = Y + (size_t)img * (64 * NT) + slot * NT;
    const float* __restrict__ yi = yr + 32 * NT;
    float zr = 0.0f, zi = 0.0f;
    for (int t = 0; t < NT; ++t) {
        float s, c;
        __sincosf(TWO_PI * (float)((kt * t) & 255) * (1.0f / 256.0f), &s, &c);
        const float a = yr[t], b = yi[t];
        zr += a * c + b * s;     // (a + ib)(c - is) real
        zi += b * c - a * s;     // imag
    }
    const size_t o = ((size_t)img * NSLOT + slot) * NKT + kt;
    Zr[o] = zr;
    Zi[o] = zi;
}

// ---------------------------------------------------------------------------
// Mix: out[o] = sum_i z[i] * w[band][i][o][kk][kt]   (complex, 32x32 per mode)
// One thread per (b, o, slot, kt) -> 262144 threads, 32-iteration i loop.
// ---------------------------------------------------------------------------
__global__ __launch_bounds__(256) void mix_kernel(const float* __restrict__ Zr,
                                                  const float* __restrict__ Zi,
                                                  const float* __restrict__ wr,
                                                  const float* __restrict__ wi,
                                                  float* __restrict__ Zmr,
                                                  float* __restrict__ Zmi) {
    const int idx  = blockIdx.x * blockDim.x + threadIdx.x;
    const int kt   = idx & 15;
    const int slot = (idx >> 4) & 31;
    const int o    = (idx >> 9) & 31;
    const int b    = idx >> 14;
    const int band = slot >> 3;
    const int kk   = slot & 7;
    float outr = 0.0f, outi = 0.0f;
    for (int i = 0; i < NC; ++i) {
        const size_t zo = (((size_t)b * NC + i) * NSLOT + slot) * NKT + kt;
        const float zr = Zr[zo], zi = Zi[zo];
        const size_t wo = ((((size_t)band * NC + i) * NC + o) * 8 + kk) * NKT + kt;
        const float a = wr[wo], c = wi[wo];
        outr += zr * a - zi * c;
        outi += zr * c + zi * a;
    }
    const size_t oo = (((size_t)b * NC + o) * NSLOT + slot) * NKT + kt;
    Zmr[oo] = outr;
    Zmi[oo] = outi;
}

// ---------------------------------------------------------------------------
// I1: inverse x-DFT (32 modes -> 256 rows) at the 16 kt columns, with 1/Nx.
// U[img][x][0..15] = Re, U[img][x][16..31] = Im  (laid out as GEMM A for I2).
// One thread per (img, x, kt) -> 2M threads, 32-iteration slot loop.
// ---------------------------------------------------------------------------
__global__ __launch_bounds__(256) void i1_kernel(const float* __restrict__ Zmr,
                                                 const float* __restrict__ Zmi,
                                                 float* __restrict__ U) {
    const int idx = blockIdx.x * blockDim.x + threadIdx.x;
    const int kt  = idx & 15;
    const int xx  = (idx >> 4) & 255;
    const int img = idx >> 12;
    float ur = 0.0f, ui = 0.0f;
    for (int s = 0; s < NSLOT; ++s) {
        const size_t zo = ((size_t)img * NSLOT + s) * NKT + kt;
        const float zr = Zmr[zo], zi = Zmi[zo];
        const int kx = kx_of_slot(s);
        float sn, cs;
        __sincosf(TWO_PI * (float)((kx * xx) & 255) * (1.0f / 256.0f), &sn, &cs);
        ur += zr * cs - zi * sn;   // e^{+i theta}
        ui += zr * sn + zi * cs;
    }
    const size_t base = ((size_t)img * NX + xx) * 32;
    U[base + kt]      = ur * (1.0f / 256.0f);
    U[base + 16 + kt] = ui * (1.0f / 256.0f);
}

// ---------------------------------------------------------------------------
// I2: per image, out[256][256] = U[256][32] x Binv[32][256]  (fp32 WMMA GEMM)
// One wave per (image, m-tile, 4-wide n-group): the A (U) fragment is loaded
// once per k-step and reused by 4 WMMAs -> U traffic cut 4x.  K = 32 -> 32
// WMMAs fully unrolled.  Binv is 32 KB and L2-resident.
// ---------------------------------------------------------------------------
__global__ __launch_bounds__(256) void i2_gemm(const float* __restrict__ U,
                                               const float* __restrict__ Binv,
                                               float* __restrict__ out) {
    const int gid  = blockIdx.x * blockDim.x + threadIdx.x;
    const int wave = gid >> 5;               // 512 * 16 * 4 = 32768 waves
    const int lane = threadIdx.x & 31;
    const int img  = wave >> 6;
    const int rest = wave & 63;
    const int m0     = (rest >> 2) << 4;     // 16 m-tiles
    const int n0base = (rest & 3) << 6;      // 4 n-groups of 64 columns

    const int mrow  = lane & 15;
    const int khalf = (lane >> 4) << 1;
    const float* __restrict__ Ua = U + ((size_t)img * NX + (m0 + mrow)) * 32;

    v8f acc[4] = {};
#pragma unroll
    for (int k0 = 0; k0 < 32; k0 += 4) {
        v2f a;
        a[0] = Ua[k0 + khalf];
        a[1] = Ua[k0 + khalf + 1];
#pragma unroll
        for (int j = 0; j < 4; ++j) {
            const int n0 = n0base + j * 16;
            v2f b;
            b[0] = Binv[(k0 + khalf)     * NT + n0 + mrow];
            b[1] = Binv[(k0 + khalf + 1) * NT + n0 + mrow];
            acc[j] = __builtin_amdgcn_wmma_f32_16x16x4_f32(
                false, a, false, b, (short)0, acc[j], false, false);
        }
    }

    float* __restrict__ O = out + (size_t)img * (NX * NT);
    const int half = lane >> 4;
#pragma unroll
    for (int j = 0; j < 4; ++j)
#pragma unroll
        for (int r = 0; r < 8; ++r)
            O[(size_t)(m0 + r + 8 * half) * NT + n0base + j * 16 + (lane & 15)] = acc[j][r];
}

// ---------------------------------------------------------------------------
// Launch: init twiddle matrices, then F1 -> F2 -> mix -> I1 -> I2 on `stream`.
// Workspace layout (floats):
//   Afwd 16384 | Binv 8192 | Y 8388608 | Zr 262144 | Zi 262144
//   Zmr 262144 | Zmi 262144 | U 4194304          (~54.6 MB total)
// ---------------------------------------------------------------------------
extern "C" void kernel_launch(void* const* d_in, const int* in_sizes, int n_in,
                              void* d_out, int out_size, void* d_ws, size_t ws_size,
                              hipStream_t stream) {
    const float* x  = (const float*)d_in[0];
    const float* wr = (const float*)d_in[1];
    const float* wi = (const float*)d_in[2];
    float* out = (float*)d_out;

    float* ws   = (float*)d_ws;
    float* Afwd = ws;
    float* Binv = Afwd + 64 * 256;
    float* Y    = Binv + 32 * 256;
    float* Zr   = Y   + (size_t)NIMG * 64 * NT;
    float* Zi   = Zr  + (size_t)NIMG * NSLOT * NKT;
    float* Zmr  = Zi  + (size_t)NIMG * NSLOT * NKT;
    float* Zmi  = Zmr + (size_t)NIMG * NSLOT * NKT;
    float* U    = Zmi + (size_t)NIMG * NSLOT * NKT;

    hipLaunchKernelGGL(init_mats, dim3(96),   dim3(256), 0, stream, Afwd, Binv);
    hipLaunchKernelGGL(f1_gemm,   dim3(1024), dim3(256), 0, stream, x, Afwd, Y);
    hipLaunchKernelGGL(f2_dft,    dim3(1024), dim3(256), 0, stream, Y, Zr, Zi);
    hipLaunchKernelGGL(mix_kernel,dim3(1024), dim3(256), 0, stream, Zr, Zi, wr, wi, Zmr, Zmi);
    hipLaunchKernelGGL(i1_kernel, dim3(8192), dim3(256), 0, stream, Zmr, Zmi, U);
    hipLaunchKernelGGL(i2_gemm,   dim3(4096), dim3(256), 0, stream, U, Binv, out);
}